// SelectSparse_1632087572808
// MI455X (gfx1250) — compile-verified
//
#include <hip/hip_runtime.h>
#include <hip/hip_bf16.h>

typedef __attribute__((ext_vector_type(2))) float v2f;
typedef __attribute__((ext_vector_type(8))) float v8f;

// ---------------------------------------------------------------------------
// Kernel 1: scores = x @ W + b via V_WMMA_F32_16X16X4_F32.
// One wave32 computes 16 node scores. Memory-bound: each lane streams 8 bytes
// of x per K-step (global_load_b64), W hits WGP$ after first touch.
// ---------------------------------------------------------------------------
__global__ __launch_bounds__(256) void select_scores_wmma(
    const float* __restrict__ x, const float* __restrict__ W,
    const float* __restrict__ bias, float* __restrict__ scores,
    int n_nodes, int channels) {
  const int lane = threadIdx.x & 31;
  const int wave_in_block = threadIdx.x >> 5;
  const long wave_id = (long)blockIdx.x * (blockDim.x >> 5) + wave_in_block;
  const long m_base = wave_id * 16;
  if (m_base >= n_nodes) return;  // wave-uniform: EXEC stays all-1s for WMMA

  const int m = (int)m_base + (lane & 15);   // A-matrix row this lane feeds
  const int koff = (lane >> 4) * 2;          // K sub-offset: lanes 0-15 -> K0,1; 16-31 -> K2,3
  const float* __restrict__ xrow = x + (long)m * channels + koff;
  const float* __restrict__ wptr = W + koff;

  v8f c = {0.f, 0.f, 0.f, 0.f, 0.f, 0.f, 0.f, 0.f};
#pragma unroll 8
  for (int k0 = 0; k0 < channels; k0 += 4) {
    v2f a = *(const v2f*)(xrow + k0);  // 8B aligned (koff even)
    v2f b = *(const v2f*)(wptr + k0);  // broadcast W across all 16 N columns
    // D = A(16x4) * B(4x16) + C ; every N column identical = dot product
    c = __builtin_amdgcn_wmma_f32_16x16x4_f32(
        /*neg_a=*/false, a, /*neg_b=*/false, b,
        /*c_mod=*/(short)0, c, /*reuse_a=*/false, /*reuse_b=*/false);
  }

  const float bv = bias[0];
  // C/D layout: VGPR r = M=r (lanes 0-15) / M=r+8 (lanes 16-31); N = lane&15.
  // Column N=0 carriers (lane 0 and lane 16) write the 16 scores.
  if ((lane & 15) == 0) {
    const int mrow = (int)m_base + ((lane >> 4) << 3);
#pragma unroll
    for (int r = 0; r < 8; ++r) scores[mrow + r] = c[r] + bv;
  }
}

// ---------------------------------------------------------------------------
// Kernel 2: per-graph top-k (k=50 of 100), descending, lowest-index ties
// first (matches jax.lax.top_k). One thread per graph; 1000 graphs total so
// this stage is negligible next to the 102 MB GEMV stream.
// Output layout (all f32, integers exact in f32):
//   [0, G*k)        node_index
//   [G*k, 2*G*k)    cluster_index
//   [2*G*k, 3*G*k)  weight
// ---------------------------------------------------------------------------
__global__ void select_topk(const float* __restrict__ scores,
                            float* __restrict__ out,
                            int num_graphs, int n_per, int k) {
  const int g = blockIdx.x * blockDim.x + threadIdx.x;
  if (g >= num_graphs) return;

  float vals[128];  // n_per <= 128
  const float* __restrict__ s = scores + (long)g * n_per;
  for (int i = 0; i < n_per; ++i) vals[i] = s[i];

  const int total = num_graphs * k;
  const int outb = g * k;
  const float NEG_INF = -__builtin_inff();
  for (int j = 0; j < k; ++j) {
    float best = NEG_INF;
    int bi = 0;
    for (int i = 0; i < n_per; ++i) {
      const float v = vals[i];
      if (v > best) { best = v; bi = i; }  // strict > => first (lowest) index wins ties
    }
    vals[bi] = NEG_INF;
    const int jg = outb + j;
    out[jg] = (float)(g * n_per + bi);      // node_index
    out[total + jg] = (float)jg;            // cluster_index
    out[2 * total + jg] = best;             // weight
  }
}

extern "C" void kernel_launch(void* const* d_in, const int* in_sizes, int n_in,
                              void* d_out, int out_size, void* d_ws, size_t ws_size,
                              hipStream_t stream) {
  const float* x = (const float*)d_in[0];
  // d_in[1] = batch (unused: segments are uniform, contiguous, sorted)
  const float* W = (const float*)d_in[2];
  const float* b = (const float*)d_in[3];
  float* out = (float*)d_out;

  const int channels = 256;
  const int n_nodes = in_sizes[0] / channels;   // 100000
  const int n_per = 100;                        // NODES_PER_GRAPH
  const int num_graphs = n_nodes / n_per;       // 1000
  const int k = (n_per + 1) / 2;                // ceil(0.5 * 100) = 50

  float* scores = (float*)d_ws;                 // n_nodes floats = 400 KB

  // GEMV: 16 nodes per wave, 8 waves per 256-thread block
  const int waves = (n_nodes + 15) / 16;
  const int blocks = (waves + 7) / 8;
  select_scores_wmma<<<blocks, 256, 0, stream>>>(x, W, b, scores, n_nodes, channels);

  // Top-k: one thread per graph
  const int tb = 256;
  select_topk<<<(num_graphs + tb - 1) / tb, tb, 0, stream>>>(scores, out,
                                                             num_graphs, n_per, k);
}